// DeformableTransformer_61272003444827
// MI455X (gfx1250) — compile-verified
//
#include <hip/hip_runtime.h>
#include <hip/hip_bf16.h>
#include <math.h>

typedef float v2f __attribute__((ext_vector_type(2)));
typedef float v8f __attribute__((ext_vector_type(8)));

// ---------------- problem constants ----------------
#define BS_      8
#define NQ_      300
#define DM_      256
#define NH_      8
#define NL_      4
#define NP_      4
#define DH_      32
#define STOT_    21760              // 128*128 + 64*64 + 32*32 + 16*16

// ---------------- GEMM: C[M,N] = A[M,K] @ W[K,N] + bias[N] --------------
// fp32 WMMA 16x16x4. Workgroup = 128 threads = 4 waves, tile 64x64, BK=32.
#define BM 64
#define BN 64
#define BK 32
#define LDSTR 36   // padded float stride for LDS rows

__global__ __launch_bounds__(128)
void gemm_wmma_f32(const float* __restrict__ A, const float* __restrict__ W,
                   const float* __restrict__ bias, float* __restrict__ C,
                   int M, int N, int K)
{
    __shared__ float sA[BM * LDSTR];   // [m][k]
    __shared__ float sB[BN * LDSTR];   // [n][k]  (B transposed)

    const int tid  = threadIdx.x;
    const int wave = tid >> 5;         // 0..3 -> M sub-tile
    const int lane = tid & 31;
    const int l16  = lane & 15;
    const int lhi  = lane >> 4;        // 0/1 -> K-pair select

    const int block_n = blockIdx.x * BN;
    const int block_m = blockIdx.y * BM;

    v8f acc[4];
    v8f z8 = {0.f,0.f,0.f,0.f,0.f,0.f,0.f,0.f};
    acc[0] = z8; acc[1] = z8; acc[2] = z8; acc[3] = z8;

    for (int k0 = 0; k0 < K; k0 += BK) {
        // ---- stage A tile: 64x32 = 512 float4, 4 per thread ----
        #pragma unroll
        for (int i = tid; i < (BM*BK)/4; i += 128) {
            int r  = i >> 3;           // 8 float4 per 32-wide row
            int c4 = (i & 7) << 2;
            int gm = block_m + r;
            float4 v;
            if (gm < M) {
                v = *(const float4*)(A + (size_t)gm * K + k0 + c4);
            } else {
                v.x = v.y = v.z = v.w = 0.f;
            }
            *(float4*)(&sA[r * LDSTR + c4]) = v;
        }
        // ---- stage B tile transposed: 32x64 = 512 float4 ----
        #pragma unroll
        for (int i = tid; i < (BK*BN)/4; i += 128) {
            int r  = i >> 4;           // k within chunk (16 float4 per 64-wide row)
            int c4 = (i & 15) << 2;    // n base
            const float4 v = *(const float4*)(W + (size_t)(k0 + r) * N + block_n + c4);
            sB[(c4 + 0) * LDSTR + r] = v.x;
            sB[(c4 + 1) * LDSTR + r] = v.y;
            sB[(c4 + 2) * LDSTR + r] = v.z;
            sB[(c4 + 3) * LDSTR + r] = v.w;
        }
        __syncthreads();

        // ---- 8 K-steps of 4, 4 N sub-tiles each -> 32 WMMAs per chunk ----
        #pragma unroll
        for (int kk = 0; kk < BK; kk += 4) {
            v2f a = *(const v2f*)(&sA[(wave * 16 + l16) * LDSTR + kk + 2 * lhi]);
            #pragma unroll
            for (int j = 0; j < 4; ++j) {
                v2f b = *(const v2f*)(&sB[(j * 16 + l16) * LDSTR + kk + 2 * lhi]);
                acc[j] = __builtin_amdgcn_wmma_f32_16x16x4_f32(
                    false, a, false, b, (short)0, acc[j], false, false);
            }
        }
        __syncthreads();
    }

    // ---- epilogue: C/D layout -> lane l16 = N col, vgpr v + 8*lhi = M row ----
    #pragma unroll
    for (int j = 0; j < 4; ++j) {
        int gn = block_n + j * 16 + l16;
        float bv = bias[gn];
        #pragma unroll
        for (int v = 0; v < 8; ++v) {
            int gm = block_m + wave * 16 + v + 8 * lhi;
            if (gm < M) C[(size_t)gm * N + gn] = acc[j][v] + bv;
        }
    }
}

// ---------------- query = tgt + query_pos ----------------
__global__ void add_kernel(const float* __restrict__ a, const float* __restrict__ b,
                           float* __restrict__ o, int n)
{
    int i = blockIdx.x * blockDim.x + threadIdx.x;
    if (i < n) o[i] = a[i] + b[i];
}

// ---------------- softmax over 16 logits + sampling locations ------------
__global__ void softmax_loc_kernel(const float* __restrict__ logits,   // [2400,128]
                                   const float* __restrict__ offsets,  // [2400,256]
                                   const float* __restrict__ refpts,   // [2400,4]
                                   const float* __restrict__ vr,       // [8,4,2]
                                   float* __restrict__ aw_out,         // [19200,16]
                                   float* __restrict__ loc_out)        // [19200,16,2]
{
    int gid = blockIdx.x * blockDim.x + threadIdx.x;   // (b*300+q)*8 + h
    if (gid >= BS_ * NQ_ * NH_) return;
    int h  = gid & 7;
    int bq = gid >> 3;
    int b  = bq / NQ_;

    // softmax
    const float* lg = logits + (size_t)bq * 128 + h * 16;
    float m = lg[0];
    #pragma unroll
    for (int i = 1; i < 16; ++i) m = fmaxf(m, lg[i]);
    float e[16], s = 0.f;
    #pragma unroll
    for (int i = 0; i < 16; ++i) { e[i] = __expf(lg[i] - m); s += e[i]; }
    float inv = 1.f / s;
    #pragma unroll
    for (int i = 0; i < 16; ++i) aw_out[(size_t)gid * 16 + i] = e[i] * inv;

    // sampling locations
    const float* rp = refpts + (size_t)bq * 4;
    #pragma unroll
    for (int l = 0; l < NL_; ++l) {
        float vx = vr[(b * NL_ + l) * 2 + 0];
        float vy = vr[(b * NL_ + l) * 2 + 1];
        float rx = rp[0] * vx, ry = rp[1] * vy;
        float rw = rp[2] * vx, rh = rp[3] * vy;
        #pragma unroll
        for (int p = 0; p < NP_; ++p) {
            float ox = offsets[(size_t)bq * 256 + h * 32 + l * 8 + p * 2 + 0];
            float oy = offsets[(size_t)bq * 256 + h * 32 + l * 8 + p * 2 + 1];
            size_t oi = ((size_t)gid * 16 + l * 4 + p) * 2;
            loc_out[oi + 0] = rx + ox * (1.0f / NP_) * rw * 0.5f;
            loc_out[oi + 1] = ry + oy * (1.0f / NP_) * rh * 0.5f;
        }
    }
}

// ---------------- bilinear sampling + attention aggregation -------------
__device__ __forceinline__ float fetchv(const float* __restrict__ vb,
                                        int ix, int iy, int W_, int H_)
{
    if (ix < 0 || ix >= W_ || iy < 0 || iy >= H_) return 0.f;
    return vb[(size_t)(iy * W_ + ix) * DM_];
}

__global__ __launch_bounds__(256)
void sample_kernel(const float* __restrict__ value,   // [8,21760,8,32]
                   const float* __restrict__ loc,     // [19200,16,2]
                   const float* __restrict__ aw,      // [19200,16]
                   float* __restrict__ attn_out)      // [2400,256]
{
    int unit = blockIdx.x * 8 + (threadIdx.x >> 5);   // one wave32 per (b,q,h)
    int d = threadIdx.x & 31;                         // channel
    if (unit >= BS_ * NQ_ * NH_) return;
    int h  = unit & 7;
    int bq = unit >> 3;
    int b  = bq / NQ_;

    const int Hs[NL_] = {128, 64, 32, 16};
    const int Ws[NL_] = {128, 64, 32, 16};
    const int S0[NL_] = {0, 16384, 20480, 21504};

    float acc = 0.f;
    #pragma unroll
    for (int l = 0; l < NL_; ++l) {
        const int H_ = Hs[l], W_ = Ws[l];
        const float* vbase = value + ((size_t)b * STOT_ + S0[l]) * DM_ + h * DH_ + d;
        #pragma unroll
        for (int p = 0; p < NP_; ++p) {
            size_t idx = (size_t)unit * 16 + l * 4 + p;
            float lx = loc[idx * 2 + 0];
            float ly = loc[idx * 2 + 1];
            float a  = aw[idx];
            float x = lx * W_ - 0.5f;
            float y = ly * H_ - 0.5f;
            float x0f = floorf(x), y0f = floorf(y);
            float wx = x - x0f, wy = y - y0f;
            int x0 = (int)x0f, y0 = (int)y0f;
            float g00 = fetchv(vbase, x0,     y0,     W_, H_);
            float g10 = fetchv(vbase, x0 + 1, y0,     W_, H_);
            float g01 = fetchv(vbase, x0,     y0 + 1, W_, H_);
            float g11 = fetchv(vbase, x0 + 1, y0 + 1, W_, H_);
            float s = g00 * (1.f - wx) * (1.f - wy) + g10 * wx * (1.f - wy)
                    + g01 * (1.f - wx) * wy         + g11 * wx * wy;
            acc += a * s;
        }
    }
    attn_out[(size_t)bq * DM_ + h * DH_ + d] = acc;
}

// ---------------- pairwise IoU + ordering ----------------
__global__ void iou_kernel(const float* __restrict__ refpts,  // [8,300,4]
                           float* __restrict__ giou_out,      // [8,300,300]
                           float* __restrict__ order_out)     // [8,300,300]
{
    int gid = blockIdx.x * blockDim.x + threadIdx.x;
    if (gid >= BS_ * NQ_ * NQ_) return;
    int j = gid % NQ_;
    int t = gid / NQ_;
    int i = t % NQ_;
    int b = t / NQ_;

    const float* ri = refpts + ((size_t)b * NQ_ + i) * 4;
    const float* rj = refpts + ((size_t)b * NQ_ + j) * 4;
    float cxi = ri[0], cyi = ri[1], wi = ri[2], hi = ri[3];
    float cxj = rj[0], cyj = rj[1], wj = rj[2], hj = rj[3];

    float x1i = cxi - 0.5f * wi, y1i = cyi - 0.5f * hi;
    float x2i = cxi + 0.5f * wi, y2i = cyi + 0.5f * hi;
    float x1j = cxj - 0.5f * wj, y1j = cyj - 0.5f * hj;
    float x2j = cxj + 0.5f * wj, y2j = cyj + 0.5f * hj;

    float ai = fmaxf(x2i - x1i, 0.f) * fmaxf(y2i - y1i, 0.f);
    float aj = fmaxf(x2j - x1j, 0.f) * fmaxf(y2j - y1j, 0.f);

    float ix = fmaxf(fminf(x2i, x2j) - fmaxf(x1i, x1j), 0.f);
    float iy = fmaxf(fminf(y2i, y2j) - fmaxf(y1i, y1j), 0.f);
    float inter = ix * iy;
    float uni = ai + aj - inter;

    giou_out[gid]  = inter / uni;
    order_out[gid] = ((cxi + cyi) > (cxj + cyj)) ? 1.f : 0.f;
}

// ---------------- launch ----------------
extern "C" void kernel_launch(void* const* d_in, const int* in_sizes, int n_in,
                              void* d_out, int out_size, void* d_ws, size_t ws_size,
                              hipStream_t stream) {
    const float* tgt        = (const float*)d_in[0];
    const float* query_pos  = (const float*)d_in[1];
    const float* refpts     = (const float*)d_in[2];
    const float* src        = (const float*)d_in[3];
    /* d_in[4] = src_spatial_shapes (compile-time constants) */
    const float* vr         = (const float*)d_in[5];
    const float* value_w    = (const float*)d_in[6];
    const float* value_b    = (const float*)d_in[7];
    const float* off_w      = (const float*)d_in[8];
    const float* off_b      = (const float*)d_in[9];
    const float* attn_w     = (const float*)d_in[10];
    const float* attn_b     = (const float*)d_in[11];
    const float* out_w      = (const float*)d_in[12];
    const float* out_b      = (const float*)d_in[13];

    float* out = (float*)d_out;
    float* ws  = (float*)d_ws;

    const size_t M_VAL = (size_t)BS_ * STOT_;     // 174080
    const size_t M_Q   = (size_t)BS_ * NQ_;       // 2400

    // workspace layout (floats; all offsets 16B-aligned)
    float* ws_value = ws;                                   // 174080*256
    float* ws_query = ws_value + M_VAL * DM_;               // 2400*256
    float* ws_off   = ws_query + M_Q * DM_;                 // 2400*256
    float* ws_attn  = ws_off   + M_Q * DM_;                 // 2400*128
    float* ws_attno = ws_attn  + M_Q * 128;                 // 2400*256

    // output layout (concatenated, return order)
    float* out_final = out;                                 // 614400
    float* out_giou  = out + 614400;                        // 720000
    float* out_order = out + 1334400;                       // 720000
    float* out_loc   = out + 2054400;                       // 614400
    float* out_aw    = out + 2668800;                       // 307200

    // 1. query = tgt + query_pos
    {
        int n = (int)(M_Q * DM_);
        add_kernel<<<(n + 255) / 256, 256, 0, stream>>>(tgt, query_pos, ws_query, n);
    }
    // 2. value = src @ value_w + value_b   (the big one: 22.8 GFLOP)
    {
        dim3 g(DM_ / BN, (unsigned)((M_VAL + BM - 1) / BM));
        gemm_wmma_f32<<<g, 128, 0, stream>>>(src, value_w, value_b, ws_value,
                                             (int)M_VAL, DM_, DM_);
    }
    // 3. offsets = query @ off_w + off_b
    {
        dim3 g(DM_ / BN, (unsigned)((M_Q + BM - 1) / BM));
        gemm_wmma_f32<<<g, 128, 0, stream>>>(ws_query, off_w, off_b, ws_off,
                                             (int)M_Q, DM_, DM_);
    }
    // 4. attn logits = query @ attn_w + attn_b
    {
        dim3 g(128 / BN, (unsigned)((M_Q + BM - 1) / BM));
        gemm_wmma_f32<<<g, 128, 0, stream>>>(ws_query, attn_w, attn_b, ws_attn,
                                             (int)M_Q, 128, DM_);
    }
    // 5. softmax + sampling locations (writes aw + loc output regions)
    {
        int n = BS_ * NQ_ * NH_;  // 19200
        softmax_loc_kernel<<<(n + 127) / 128, 128, 0, stream>>>(
            ws_attn, ws_off, refpts, vr, out_aw, out_loc);
    }
    // 6. bilinear sampling + aggregation (one wave32 per (b,q,h))
    {
        sample_kernel<<<(BS_ * NQ_ * NH_) / 8, 256, 0, stream>>>(
            ws_value, out_loc, out_aw, ws_attno);
    }
    // 7. out = attn_out @ out_w + out_b
    {
        dim3 g(DM_ / BN, (unsigned)((M_Q + BM - 1) / BM));
        gemm_wmma_f32<<<g, 128, 0, stream>>>(ws_attno, out_w, out_b, out_final,
                                             (int)M_Q, DM_, DM_);
    }
    // 8. pairwise IoU + order
    {
        int n = BS_ * NQ_ * NQ_;  // 720000
        iou_kernel<<<(n + 255) / 256, 256, 0, stream>>>(refpts, out_giou, out_order);
    }
}